// DeformableConvLayer_change_pred_39187281609286
// MI455X (gfx1250) — compile-verified
//
#include <hip/hip_runtime.h>

// Problem constants (match reference)
#define Bc   2
#define Hc   48
#define Wc   48
#define Cc   32
#define Uc   24
#define KPTS 9                    // KH*KW
#define Nn   (KPTS * Uc)          // 216
#define Npad 224                  // N padded to 16-multiple
#define Mrows (Bc * Hc * Wc)      // 4608

typedef __attribute__((ext_vector_type(2))) float v2f;
typedef __attribute__((ext_vector_type(8))) float v8f;

// ---------------------------------------------------------------------------
// Stage 1: D[b,y,x,n] = sum_c x_shift[b,y,x,c] * K[c,n]  (f32 GEMM via WMMA)
//   x_shift[b,y,x,c] = inputs[b,y-1,x-1,c] (0 outside)   [the reference's
//   padded-map-with-unpadded-coords gather]
//   K[c, n] with n = ij*U + u  -> kernel[ij*(U*C) + u*C + c]
// One wave computes one 16x16 tile; K-loop = 8 x v_wmma_f32_16x16x4_f32.
// ---------------------------------------------------------------------------
__global__ void deform_stage1_gemm(const float* __restrict__ x,
                                   const float* __restrict__ kern,
                                   float* __restrict__ D) {
    const int wave  = (blockIdx.x * blockDim.x + threadIdx.x) >> 5;
    const int lane  = threadIdx.x & 31;
    const int lm    = lane & 15;
    const int khalf = lane >> 4;          // 0: K pair {k0,k0+1}; 1: {k0+2,k0+3}

    const int nTilesN = Npad / 16;        // 14
    const int mt = wave / nTilesN;        // 0..287
    const int nt = wave % nTilesN;        // 0..13

    // ---- A operand row: decode (b, y, x) from m; shifted sample (y-1, x-1)
    const int m   = mt * 16 + lm;
    const int b   = m / (Hc * Wc);
    const int rem = m - b * (Hc * Wc);
    const int yy  = rem / Wc;
    const int xx  = rem - yy * Wc;
    const int ys  = yy - 1, xs = xx - 1;
    const bool mvalid = (ys >= 0) && (xs >= 0) && (ys < Hc) && (xs < Wc);
    const int ysc = mvalid ? ys : 0, xsc = mvalid ? xs : 0;
    const float* xrow = x + (((size_t)b * Hc + ysc) * Wc + xsc) * Cc;

    // ---- B operand column: n -> (ij, u); channels are innermost in kernel
    const int n = nt * 16 + lm;
    const bool nvalid = (n < Nn);
    const int ij = nvalid ? (n / Uc) : 0;
    const int u  = nvalid ? (n - ij * Uc) : 0;
    const float* krow = kern + (size_t)ij * (Uc * Cc) + (size_t)u * Cc;

    v8f acc = {0.f, 0.f, 0.f, 0.f, 0.f, 0.f, 0.f, 0.f};

#pragma unroll
    for (int k0 = 0; k0 < Cc; k0 += 4) {
        const int c = k0 + 2 * khalf;
        // A fragment (16x4 f32): VGPR0 = K=c, VGPR1 = K=c+1 (per ISA layout)
        float ax = xrow[c];
        float ay = xrow[c + 1];
        v2f a;  a.x = mvalid ? ax : 0.f;  a.y = mvalid ? ay : 0.f;
        // B fragment (4x16 f32): same K striping across lane halves
        float bx = krow[c];
        float by = krow[c + 1];
        v2f bb; bb.x = nvalid ? bx : 0.f; bb.y = nvalid ? by : 0.f;

        acc = __builtin_amdgcn_wmma_f32_16x16x4_f32(
            /*neg_a=*/false, a, /*neg_b=*/false, bb,
            /*c_mod=*/(short)0, acc, /*reuse_a=*/false, /*reuse_b=*/false);
    }

    // ---- store C/D: VGPR r -> M = r + 8*khalf, N = lm
    float* Dbase = D + (size_t)(mt * 16) * Npad + (size_t)(nt * 16 + lm);
#pragma unroll
    for (int r = 0; r < 8; ++r) {
        const int mm = r + 8 * khalf;
        Dbase[(size_t)mm * Npad] = acc[r];
    }
}

// ---------------------------------------------------------------------------
// Stage 2: per-output scalar bilinear gather from D + bias.
// out[b,h,w,u] = sum_ij bilerp(D[b,:,:, ij*U+u]; y,x) + bias[u]
// ---------------------------------------------------------------------------
__global__ void deform_stage2_sample(const float* __restrict__ off,
                                     const float* __restrict__ D,
                                     const float* __restrict__ bias,
                                     float* __restrict__ out) {
    const int idx = blockIdx.x * blockDim.x + threadIdx.x;
    if (idx >= Bc * Hc * Wc * Uc) return;

    const int u = idx % Uc;
    const int p = idx / Uc;           // (b*H + h)*W + w
    const int w = p % Wc;
    const int t = p / Wc;
    const int h = t % Hc;
    const int b = t / Hc;

    const float* offp = off + (size_t)p * Nn * 2;
    const float* Db   = D + (size_t)b * (Hc * Wc) * Npad;

    float acc = 0.f;
#pragma unroll
    for (int ij = 0; ij < KPTS; ++ij) {
        const int i = ij / 3, j = ij % 3;
        const int hh = h + i - 1;
        const int ww = w + j - 1;
        const bool valid = (hh >= 0) && (hh < Hc) && (ww >= 0) && (ww < Wc);
        const float yb = valid ? (float)hh : 0.f;
        const float xb = valid ? (float)ww : 0.f;

        const int k = ij * Uc + u;
        const float dy = offp[2 * k + 0];
        const float dx = offp[2 * k + 1];

        const float yf = fminf(fmaxf(yb + dy, 0.f), (float)(Hc - 1));
        const float xf = fminf(fmaxf(xb + dx, 0.f), (float)(Wc - 1));

        int y0 = (int)floorf(yf); y0 = min(max(y0, 0), Hc - 1);
        int x0 = (int)floorf(xf); x0 = min(max(x0, 0), Wc - 1);
        const int y1 = min(y0 + 1, Hc - 1);
        const int x1 = min(x0 + 1, Wc - 1);

        const float y0f = (float)y0, y1f = (float)y1;
        const float x0f = (float)x0, x1f = (float)x1;
        const float w0 = (y1f - yf) * (x1f - xf);
        const float w1 = (y1f - yf) * (xf - x0f);
        const float w2 = (yf - y0f) * (x1f - xf);
        const float w3 = (yf - y0f) * (xf - x0f);

        const float d00 = Db[((size_t)y0 * Wc + x0) * Npad + k];
        const float d01 = Db[((size_t)y0 * Wc + x1) * Npad + k];
        const float d10 = Db[((size_t)y1 * Wc + x0) * Npad + k];
        const float d11 = Db[((size_t)y1 * Wc + x1) * Npad + k];

        acc += w0 * d00 + w1 * d01 + w2 * d10 + w3 * d11;
    }
    out[idx] = acc + bias[u];
}

// ---------------------------------------------------------------------------
// Fallback (only if d_ws is too small for D): fused naive gather+dot.
// ---------------------------------------------------------------------------
__global__ void deform_fused_naive(const float* __restrict__ x,
                                   const float* __restrict__ off,
                                   const float* __restrict__ kern,
                                   const float* __restrict__ bias,
                                   float* __restrict__ out) {
    const int idx = blockIdx.x * blockDim.x + threadIdx.x;
    if (idx >= Bc * Hc * Wc * Uc) return;

    const int u = idx % Uc;
    const int p = idx / Uc;
    const int w = p % Wc;
    const int t = p / Wc;
    const int h = t % Hc;
    const int b = t / Hc;

    const float* offp = off + (size_t)p * Nn * 2;
    float acc = 0.f;
    for (int ij = 0; ij < KPTS; ++ij) {
        const int i = ij / 3, j = ij % 3;
        const int hh = h + i - 1;
        const int ww = w + j - 1;
        const bool valid = (hh >= 0) && (hh < Hc) && (ww >= 0) && (ww < Wc);
        const float yb = valid ? (float)hh : 0.f;
        const float xb = valid ? (float)ww : 0.f;

        const int k = ij * Uc + u;
        const float yf = fminf(fmaxf(yb + offp[2 * k + 0], 0.f), (float)(Hc - 1));
        const float xf = fminf(fmaxf(xb + offp[2 * k + 1], 0.f), (float)(Wc - 1));
        int y0 = (int)floorf(yf); y0 = min(max(y0, 0), Hc - 1);
        int x0 = (int)floorf(xf); x0 = min(max(x0, 0), Wc - 1);
        const int y1 = min(y0 + 1, Hc - 1);
        const int x1 = min(x0 + 1, Wc - 1);

        const float wgt[4] = {
            ((float)y1 - yf) * ((float)x1 - xf),
            ((float)y1 - yf) * (xf - (float)x0),
            (yf - (float)y0) * ((float)x1 - xf),
            (yf - (float)y0) * (xf - (float)x0)};
        const int yc[4] = {y0, y0, y1, y1};
        const int xc[4] = {x0, x1, x0, x1};

        const float* kp = kern + (size_t)ij * (Uc * Cc) + (size_t)u * Cc;
        for (int q = 0; q < 4; ++q) {
            const int ys = yc[q] - 1, xs = xc[q] - 1;
            float dot = 0.f;
            if (ys >= 0 && xs >= 0 && ys < Hc && xs < Wc) {
                const float* xr = x + (((size_t)b * Hc + ys) * Wc + xs) * Cc;
                for (int c = 0; c < Cc; ++c) dot += xr[c] * kp[c];
            }
            acc += wgt[q] * dot;
        }
    }
    out[idx] = acc + bias[u];
}

extern "C" void kernel_launch(void* const* d_in, const int* in_sizes, int n_in,
                              void* d_out, int out_size, void* d_ws, size_t ws_size,
                              hipStream_t stream) {
    const float* inputs = (const float*)d_in[0];
    const float* offset = (const float*)d_in[1];
    const float* kernel = (const float*)d_in[2];
    const float* bias   = (const float*)d_in[3];
    float* out = (float*)d_out;

    const int nout = Bc * Hc * Wc * Uc;               // 110592
    const size_t needD = (size_t)Mrows * Npad * sizeof(float);  // ~4.1 MB

    if (ws_size >= needD) {
        float* D = (float*)d_ws;
        // 288 M-tiles * 14 N-tiles = 4032 waves; 8 waves / 256-thread block
        const int waves   = (Mrows / 16) * (Npad / 16);
        const int threads = waves * 32;               // 129024
        deform_stage1_gemm<<<threads / 256, 256, 0, stream>>>(inputs, kernel, D);
        deform_stage2_sample<<<(nout + 255) / 256, 256, 0, stream>>>(offset, D, bias, out);
    } else {
        deform_fused_naive<<<(nout + 255) / 256, 256, 0, stream>>>(inputs, offset, kernel, bias, out);
    }
}